// WaveFront_46626164965542
// MI455X (gfx1250) — compile-verified
//
#include <hip/hip_runtime.h>
#include <hip/hip_bf16.h>

typedef __attribute__((ext_vector_type(2))) float v2f;
typedef __attribute__((ext_vector_type(8))) float v8f;

#define SYNC_BLOCK() do { __threadfence_block(); __syncthreads(); } while (0)

__device__ __forceinline__ float geluf(float x) {
    // exact GELU: x * 0.5 * (1 + erf(x/sqrt(2)))
    return 0.5f * x * (1.0f + erff(x * 0.70710678118654752f));
}
__device__ __forceinline__ float sigmoidf_(float z) {
    return 1.0f / (1.0f + __expf(-z));
}

// ---------------------------------------------------------------------------
// Kernel A: front detector (non-MLP part).
// Reproduces jax.lax.top_k(-score) semantics: coords are ascending, so the 8
// selected fronts are the first 8 discontinuity indices; if fewer than 8,
// padding slots are the first non-disc indices (stable -inf tie-break).
// ---------------------------------------------------------------------------
__global__ __launch_bounds__(256) void front_kernel(const float* __restrict__ x_ic,
                                                    float* __restrict__ fp_out,
                                                    int* __restrict__ count_out,
                                                    float* __restrict__ uLws,
                                                    float* __restrict__ uRws) {
    const int X = 1024;
    const int b = blockIdx.x;
    const float* den = x_ic + (size_t)(b * 3 + 0) * X;
    const float* crd = x_ic + (size_t)(b * 3 + 2) * X;

    __shared__ unsigned char disc[1023];
    __shared__ int vidx[8];
    __shared__ int pidx[8];
    __shared__ int nvalid_s;

    for (int j = threadIdx.x; j < X - 1; j += 256)
        disc[j] = (fabsf(den[j] - den[j + 1]) > 1e-6f) ? 1 : 0;
    __syncthreads();

    if (threadIdx.x == 0) {
        int nd = 0, nv = 0, np = 0;
        for (int j = 0; j < X - 1; ++j) {
            if (disc[j]) { if (nv < 8) vidx[nv++] = j; ++nd; }
            else         { if (np < 8) pidx[np++] = j; }
        }
        nvalid_s = nv;
        count_out[b] = nd;          // int32 bits into output buffer
    }
    __syncthreads();

    if (threadIdx.x < 8) {
        const int k  = threadIdx.x;
        const int nv = nvalid_s;
        const int j  = (k < nv) ? vidx[k] : pidx[k - nv];
        const float uL = den[j];
        const float uR = den[j + 1];
        const float fc = 0.5f * (crd[j] + crd[j + 1]);
        const float iv = disc[j] ? 1.0f : 0.0f;
        fp_out[(b * 6 + 2) * 8 + k] = uL;
        fp_out[(b * 6 + 3) * 8 + k] = uR;
        fp_out[(b * 6 + 4) * 8 + k] = fc;
        fp_out[(b * 6 + 5) * 8 + k] = iv;
        uLws[b * 8 + k] = uL;
        uRws[b * 8 + k] = uR;
    }
}

// ---------------------------------------------------------------------------
// Kernel B: speed-predictor MLP on 64 samples, GEMMs via V_WMMA_F32_16X16X4_F32.
// f32 WMMA keeps exact f32 precision (tiny K-step of 4, 16x16 tiles).
// Layouts per CDNA5 ISA 7.12.2:
//   A 16x4 f32 : lanes 0-15 -> K=0(v0),K=1(v1); lanes 16-31 -> K=2(v0),K=3(v1)
//   B 4x16 f32 : v0: lanes0-15 row K=0, lanes16-31 row K=2; v1: K=1 / K=3
//   C/D 16x16  : VGPR i: lanes0-15 M=i, lanes16-31 M=i+8; N = lane&15
// Wave id is readfirstlane'd so the tile loop is a scalar (SALU) loop and
// EXEC stays all ones around every WMMA issue.
// ---------------------------------------------------------------------------
template <int M, int N, int K>
__device__ __forceinline__ void gemm_wmma(const float* __restrict__ A,   // MxK row-major
                                          const float* __restrict__ B,   // KxN row-major
                                          const float* __restrict__ bias,// N
                                          const float* __restrict__ Cadd,// optional MxN
                                          float* __restrict__ D,         // MxN
                                          int tid) {
    const int wave = __builtin_amdgcn_readfirstlane(tid >> 5);  // uniform
    const int lane = tid & 31;
    const int half = lane >> 4;   // 0: lanes 0-15, 1: lanes 16-31
    const int l    = lane & 15;
    constexpr int MT = M / 16, NT = N / 16;
    for (int t = wave; t < MT * NT; t += 8) {
        const int tm = t / NT, tn = t % NT;
        const float* Ab = A + (size_t)tm * 16 * K;
        const float* Bb = B + tn * 16;
        v8f c = {0.f, 0.f, 0.f, 0.f, 0.f, 0.f, 0.f, 0.f};
#pragma unroll 4
        for (int k0 = 0; k0 < K; k0 += 4) {
            const int ka = k0 + 2 * half;
            v2f a, bb;
            a[0]  = Ab[(size_t)l * K + ka];
            a[1]  = Ab[(size_t)l * K + ka + 1];
            bb[0] = Bb[(size_t)ka * N + l];
            bb[1] = Bb[(size_t)(ka + 1) * N + l];
            c = __builtin_amdgcn_wmma_f32_16x16x4_f32(false, a, false, bb,
                                                      (short)0, c, false, false);
        }
#pragma unroll
        for (int i = 0; i < 8; ++i) {
            const int row = tm * 16 + (half << 3) + i;
            const int col = tn * 16 + l;
            float v = c[i] + bias[col];
            if (Cadd) v += Cadd[(size_t)row * N + col];
            D[(size_t)row * N + col] = v;
        }
    }
}

// LayerNorm over 64 rows x 128 features; tid<64 is wave-uniform (2 full waves).
__device__ __forceinline__ void ln_rows(const float* __restrict__ X,
                                        float* __restrict__ Y,
                                        const float* __restrict__ g,
                                        const float* __restrict__ b, int tid) {
    if (tid < 64) {
        const float* xr = X + (size_t)tid * 128;
        float m = 0.f;
        for (int j = 0; j < 128; ++j) m += xr[j];
        m *= (1.0f / 128.0f);
        float v = 0.f;
        for (int j = 0; j < 128; ++j) { const float d = xr[j] - m; v += d * d; }
        v *= (1.0f / 128.0f);
        const float inv = rsqrtf(v + 1e-5f);
        float* yr = Y + (size_t)tid * 128;
        for (int j = 0; j < 128; ++j) yr[j] = (xr[j] - m) * inv * g[j] + b[j];
    }
}

struct MLPParams {
    const float *W_in, *b_in, *g_in, *be_in;
    const float *rg[3], *rb[3], *W1[3], *rb1[3], *W2[3], *rb2[3];
    const float *g_o, *be_o, *Wh1, *bh1, *Wh2, *bh2;
};

__global__ __launch_bounds__(256) void mlp_wmma_kernel(MLPParams P,
                                                       float* __restrict__ ws,
                                                       float* __restrict__ fp_out) {
    const int tid = (int)threadIdx.x;
    float* uL = ws;             // 64
    float* uR = ws + 64;        // 64
    float* fe = ws + 128;       // 64 x 8   (K padded 6->8)
    float* Wp = ws + 640;       // 8 x 128  (W_in zero-padded)
    float* h  = ws + 1664;      // 64 x 128
    float* tb = ws + 9856;      // 64 x 128 scratch
    float* t1 = ws + 18048;     // 64 x 256
    float* o1 = ws + 34432;     // 64 x 64

    if (tid < 64) {
        const float a = uL[tid], q = uR[tid];
        const float d = a - q;
        float* f = fe + tid * 8;
        f[0] = a; f[1] = q; f[2] = d; f[3] = fabsf(d);
        f[4] = 0.5f * (a + q);
        f[5] = (d > 0.f) ? 1.f : ((d < 0.f) ? -1.f : 0.f);
        f[6] = 0.f; f[7] = 0.f;
    }
    for (int i = tid; i < 8 * 128; i += 256) {
        const int r = i >> 7, c = i & 127;
        Wp[i] = (r < 6) ? P.W_in[r * 128 + c] : 0.f;
    }
    SYNC_BLOCK();

    // h_pre = feats @ W_in + b_in ; h = gelu(LN(h_pre))
    gemm_wmma<64, 128, 8>(fe, Wp, P.b_in, nullptr, h, tid);
    SYNC_BLOCK();
    ln_rows(h, tb, P.g_in, P.be_in, tid);
    SYNC_BLOCK();
    for (int i = tid; i < 64 * 128; i += 256) h[i] = geluf(tb[i]);
    SYNC_BLOCK();

    // 3 residual blocks: h += gelu(LN(h)@W1 + b1) @ W2 + b2
    for (int r = 0; r < 3; ++r) {
        ln_rows(h, tb, P.rg[r], P.rb[r], tid);
        SYNC_BLOCK();
        gemm_wmma<64, 256, 128>(tb, P.W1[r], P.rb1[r], nullptr, t1, tid);
        SYNC_BLOCK();
        for (int i = tid; i < 64 * 256; i += 256) t1[i] = geluf(t1[i]);
        SYNC_BLOCK();
        gemm_wmma<64, 128, 256>(t1, P.W2[r], P.rb2[r], /*Cadd=*/h, /*D=*/h, tid);
        SYNC_BLOCK();
    }

    // head: o = LN(h); o1 = gelu(o@Wh1+bh1); out = o1@Wh2+bh2
    ln_rows(h, tb, P.g_o, P.be_o, tid);
    SYNC_BLOCK();
    gemm_wmma<64, 64, 128>(tb, P.Wh1, P.bh1, nullptr, o1, tid);
    SYNC_BLOCK();
    for (int i = tid; i < 64 * 64; i += 256) o1[i] = geluf(o1[i]);
    SYNC_BLOCK();

    if (tid < 64) {
        const float* orow = o1 + (size_t)tid * 64;
        float s0 = P.bh2[0], s1 = P.bh2[1];
        for (int j = 0; j < 64; ++j) {
            s0 += orow[j] * P.Wh2[j * 2 + 0];
            s1 += orow[j] * P.Wh2[j * 2 + 1];
        }
        const int b = tid >> 3, k = tid & 7;
        fp_out[(b * 6 + 0) * 8 + k] = s0;   // speed ch 0
        fp_out[(b * 6 + 1) * 8 + k] = s1;   // speed ch 1
    }
}

// ---------------------------------------------------------------------------
// Kernel C: region indexer over B*T*X = 2.1M points (bandwidth-critical).
// One block per (b,t) row: per-front weights/positions are block-uniform and
// computed once into LDS. Each thread handles 4 consecutive x with float4
// loads and int4/float4 stores (B128 vmem ops).
// ---------------------------------------------------------------------------
__global__ __launch_bounds__(256) void region_kernel(const float* __restrict__ xq,
                                                     const float* __restrict__ fp,
                                                     int* __restrict__ ridx,
                                                     float* __restrict__ irs) {
    const int T = 256, X = 1024;
    const float SHARP = 50.0f;
    const int bt = blockIdx.x;          // 1 block per (b,t)
    const int b  = bt >> 8;             // T = 256
    const int t  = bt & 255;

    __shared__ float s_p1[8], s_p2[8], s_shock2[8], s_rare[8];

    if (threadIdx.x < 8) {
        const int k = threadIdx.x;
        const float tu = xq[((size_t)(b * 3 + 1) * T + t) * X];      // channel -2, x=0
        const float f1 = fp[(b * 6 + 0) * 8 + k];
        const float f2 = fp[(b * 6 + 1) * 8 + k];
        const float fc = fp[(b * 6 + 4) * 8 + k];
        const float vv = fp[(b * 6 + 5) * 8 + k];
        s_rare[k]   = sigmoidf_( SHARP * f2) * vv;
        s_shock2[k] = 2.0f * sigmoidf_(-SHARP * f2) * vv;
        s_p1[k] = fc + f1 * tu;
        s_p2[k] = fc + f2 * tu;
    }
    __syncthreads();

    const int x0 = (int)threadIdx.x * 4;
    const float4 xv4 =
        *(const float4*)(xq + ((size_t)(b * 3 + 2) * T + t) * X + x0);  // channel -1
    float xv[4] = {xv4.x, xv4.y, xv4.z, xv4.w};
    float acc[4] = {0.f, 0.f, 0.f, 0.f};

#pragma unroll
    for (int k = 0; k < 8; ++k) {
        const float p1 = s_p1[k], p2 = s_p2[k];
        const float sw = s_shock2[k], rw = s_rare[k];
#pragma unroll
        for (int i = 0; i < 4; ++i) {
            const float r1 = sigmoidf_(SHARP * (xv[i] - p1));
            const float r2 = sigmoidf_(SHARP * (xv[i] - p2));
            acc[i] += r1 * sw + (r1 + r2) * rw;
        }
    }

    const size_t o = ((size_t)b * T + t) * X + x0;
    int4 ri;
    float4 ir;
    ri.x = (int)rintf(acc[0]);  ir.x = 0.5f - 0.5f * cospif(acc[0]);
    ri.y = (int)rintf(acc[1]);  ir.y = 0.5f - 0.5f * cospif(acc[1]);
    ri.z = (int)rintf(acc[2]);  ir.z = 0.5f - 0.5f * cospif(acc[2]);
    ri.w = (int)rintf(acc[3]);  ir.w = 0.5f - 0.5f * cospif(acc[3]);
    *(int4*)(ridx + o)  = ri;
    *(float4*)(irs + o) = ir;
}

// ---------------------------------------------------------------------------
// Host launcher. Input order = setup_inputs() dict insertion order, params
// flattened depth-first: x_ic, x_query, W_in, b_in, g_in, be_in,
// res[i]{g,b,W1,b1,W2,b2} for i=0..2, g_o, be_o, Wh1, bh1, Wh2, bh2.
// Output = concat(front_params[8,6,8] f32, front_count[8] i32,
//                 region_idx[8,256,1024] i32, is_riemann_soft[...] f32).
// ---------------------------------------------------------------------------
extern "C" void kernel_launch(void* const* d_in, const int* in_sizes, int n_in,
                              void* d_out, int out_size, void* d_ws, size_t ws_size,
                              hipStream_t stream) {
    const float* x_ic = (const float*)d_in[0];
    const float* x_q  = (const float*)d_in[1];

    MLPParams P;
    int i = 2;
    P.W_in  = (const float*)d_in[i++];
    P.b_in  = (const float*)d_in[i++];
    P.g_in  = (const float*)d_in[i++];
    P.be_in = (const float*)d_in[i++];
    for (int r = 0; r < 3; ++r) {
        P.rg[r]  = (const float*)d_in[i++];
        P.rb[r]  = (const float*)d_in[i++];
        P.W1[r]  = (const float*)d_in[i++];
        P.rb1[r] = (const float*)d_in[i++];
        P.W2[r]  = (const float*)d_in[i++];
        P.rb2[r] = (const float*)d_in[i++];
    }
    P.g_o  = (const float*)d_in[i++];
    P.be_o = (const float*)d_in[i++];
    P.Wh1  = (const float*)d_in[i++];
    P.bh1  = (const float*)d_in[i++];
    P.Wh2  = (const float*)d_in[i++];
    P.bh2  = (const float*)d_in[i++];

    float* out = (float*)d_out;
    float* fp   = out;                        // 8*6*8 = 384
    int*   cnt  = (int*)(out + 384);          // 8
    int*   ridx = (int*)(out + 392);          // 2097152
    float* irs  = out + 392 + 2097152;        // 2097152
    float* ws   = (float*)d_ws;

    front_kernel<<<8, 256, 0, stream>>>(x_ic, fp, cnt, ws, ws + 64);
    mlp_wmma_kernel<<<1, 256, 0, stream>>>(P, ws, fp);
    region_kernel<<<2048, 256, 0, stream>>>(x_q, fp, ridx, irs);
}